// ModulateConv2d_45638322487531
// MI455X (gfx1250) — compile-verified
//
#include <hip/hip_runtime.h>
#include <cstdint>

// ModulateConv2d permute+scale:
//   out[((b*L + ho*64+wo)*64 + c)*25 + (i*5+jj)] =
//       x[b, c, ho*5+i, wo*5+jj] * mask[b, i*5+jj, ho, wo]
// K==STRIDE==5 -> non-overlapping tiling, pure bandwidth problem (~423 MB
// @ 23.3 TB/s ~= 18 us). No contraction dim -> WMMA inapplicable; optimize
// the CDNA5 data path: async global->LDS staging (ASYNCcnt) + coalesced b128
// stores, with incremental index arithmetic to keep VALU off the critical path.

#define BDIM 320      // 10 wave32s
#define TW   8        // wo tiles per workgroup

static constexpr int Cc  = 64;
static constexpr int Hh  = 320;
static constexpr int Ww  = 320;
static constexpr int K2  = 25;
static constexpr int HO  = 64;
static constexpr int WO  = 64;
static constexpr int LL  = HO * WO;          // 4096
static constexpr int XT_FLOATS = Cc * 5 * (TW * 5);   // 12800 floats = 50 KB
static constexpr int MT_FLOATS = K2 * TW;             // 200 floats

// ---- CDNA5 async global->LDS copy (ASYNCcnt path) -------------------------
typedef int v4i __attribute__((ext_vector_type(4)));
typedef __attribute__((address_space(1))) v4i v4i_g;
typedef __attribute__((address_space(3))) v4i v4i_l;

__device__ __forceinline__ void async_copy_b128(const float* g, float* l) {
#if __has_builtin(__builtin_amdgcn_global_load_async_to_lds_b128)
    __builtin_amdgcn_global_load_async_to_lds_b128(
        (v4i_g*)g, (v4i_l*)l, /*imm offset=*/0, /*cpol=*/0);
#else
    uint32_t loff = (uint32_t)(uintptr_t)l;
    asm volatile("global_load_async_to_lds_b128 %0, %1, off"
                 :: "v"(loff), "v"(g) : "memory");
#endif
}

__device__ __forceinline__ void wait_async_zero() {
#if __has_builtin(__builtin_amdgcn_s_wait_asynccnt)
    __builtin_amdgcn_s_wait_asynccnt(0);
#else
    asm volatile("s_wait_asynccnt 0" ::: "memory");
#endif
}

// ---------------------------------------------------------------------------
__global__ __launch_bounds__(BDIM)
void modconv_permute_kernel(const float* __restrict__ x,
                            const float* __restrict__ mask,
                            float* __restrict__ out) {
    // xt layout: [c][i][col], col in [0, TW*5) ; stride(c)=200, stride(i)=40
    __shared__ float xt[XT_FLOATS];
    __shared__ float mt[MT_FLOATS];   // [j][wo_off]

    const int tid = threadIdx.x;
    const int wo0 = blockIdx.x * TW;  // 0..56 step 8
    const int ho  = blockIdx.y;       // 0..63
    const int b   = blockIdx.z;       // 0..7

    // ---- stage mask tile: mask[b, j, ho, wo0+w] ----
    if (tid < MT_FLOATS) {
        const int j = tid / TW;
        const int w = tid - j * TW;
        mt[tid] = mask[((size_t)b * K2 + j) * LL + ho * WO + wo0 + w];
    }

    // ---- async stage x tile: 3200 b128 transfers, all 16B-aligned ----
    // float4 index idx = tid + iter*320; seg = idx/10 = tid/10 + 32*iter,
    // q4 = tid%10 (loop-invariant). seg = c*5 + i; advance c,i incrementally.
    {
        const int q4 = tid % 10;            // one div/mod chain at entry
        const int s0 = tid / 10;            // 0..31
        int c = s0 / 5;                     // tiny constant divs, entry only
        int i = s0 - c * 5;
        // LDS offset advances uniformly by 1280 floats/iter (5*40 == 200,
        // so the i->c wrap is address-neutral).
        float* ldst = &xt[s0 * 40 + q4 * 4];
        const float* xbase = x + (size_t)b * Cc * Hh * Ww
                               + (size_t)(ho * 5) * Ww + wo0 * 5 + q4 * 4;
        #pragma unroll
        for (int iter = 0; iter < 10; ++iter) {
            const float* gsrc = xbase + (size_t)c * (Hh * Ww) + i * Ww;
            async_copy_b128(gsrc, ldst);
            // seg += 32  ==>  c += 6, i += 2, wrap i
            c += 6; i += 2;
            if (i >= 5) { i -= 5; c += 1; }
            ldst += 1280;
        }
    }

    wait_async_zero();     // this wave's async copies have landed in LDS
    __syncthreads();       // make all waves' LDS writes visible

    // ---- permute + scale out of LDS; output region is contiguous ----
    // element e = tid*4 + 1280*iter; e never crosses a 1600 (wo tile)
    // boundary inside a float4 (1600 % 4 == 0). Maintain
    //   wo_off = e/1600, r = e%1600, c0 = r/25, j0 = r%25
    // incrementally: r += 1280 => c0 += 51, j0 += 5 (wraps handled below;
    // the 1600-wrap subtracts exactly 64 from c0 since 1600/25 == 64).
    float* ob = out + ((size_t)b * LL + (size_t)ho * WO + wo0) * (size_t)(Cc * K2);
    {
        int wo_off = 0;
        int r  = tid * 4;                   // 0..1276 (< 1600)
        int c0 = r / 25;                    // one constant-div chain at entry
        int j0 = r - c0 * 25;
        #pragma unroll
        for (int iter = 0; iter < 10; ++iter) {
            const int e = tid * 4 + iter * 1280;
            const int woff5 = wo_off * 5;
            float4 v;
            float* vp = &v.x;
            #pragma unroll
            for (int q = 0; q < 4; ++q) {
                int j = j0 + q;
                int c = c0;
                if (j >= 25) { j -= 25; c += 1; }
                const int i = (j * 52) >> 8;            // j/5 for j in [0,25)
                // xt[c][i][woff5 + (j - 5i)] = xt[c*200 + woff5 + j + 35*i]
                vp[q] = xt[c * 200 + woff5 + j + 35 * i] * mt[j * TW + wo_off];
            }
            *(float4*)(ob + e) = v;                     // coalesced b128 store
            // advance e by 1280
            r += 1280; c0 += 51; j0 += 5;
            if (j0 >= 25) { j0 -= 25; c0 += 1; }
            if (r >= 1600) { r -= 1600; wo_off += 1; c0 -= 64; }
        }
    }
}

extern "C" void kernel_launch(void* const* d_in, const int* in_sizes, int n_in,
                              void* d_out, int out_size, void* d_ws, size_t ws_size,
                              hipStream_t stream) {
    const float* x    = (const float*)d_in[0];   // [8, 64, 320, 320] f32
    const float* mask = (const float*)d_in[1];   // [8, 25, 64, 64]  f32
    float* out = (float*)d_out;                  // [8, 4096*64, 25] f32

    dim3 grid(WO / TW, HO, 8);   // (8, 64, 8) = 4096 workgroups
    modconv_permute_kernel<<<grid, BDIM, 0, stream>>>(x, mask, out);
}